// decoder_63823214019242
// MI455X (gfx1250) — compile-verified
//
#include <hip/hip_runtime.h>
#include <stdint.h>

#define S_DIM 512
#define TILE  32      // 32x32 output tile per workgroup (both p planes)
#define KC    64      // K chunk staged per double-buffer phase
#define APAD  68      // A row padded 64 -> 68 floats (bank-conflict-free, 16B aligned rows)

// Low 32 bits of a generic pointer to __shared__ == LDS byte offset (addrspacecast 3->0
// puts the shared-aperture base in the high 32 bits only).
__device__ __forceinline__ uint32_t lds_off(const void* p) {
  return (uint32_t)(uintptr_t)p;
}

// CDNA5 async global->LDS copy: per-lane 16B, tracked by ASYNCcnt (not LOADcnt).
__device__ __forceinline__ void async_ld_b128(uint32_t lds, const void* g) {
  asm volatile("global_load_async_to_lds_b128 %0, %1, off"
               :: "v"(lds), "v"(g)
               : "memory");
}

__global__ __launch_bounds__(256)
void fuzzy_maxmin_decoder(const float* __restrict__ t, float* __restrict__ out) {
  // Double-buffered tiles: A = t0[s, kchunk] (padded), B0 = t0[kchunk, o], B1 = t1[kchunk, o]
  __shared__ float Ash [2][TILE][APAD];
  __shared__ float B0sh[2][KC][TILE];
  __shared__ float B1sh[2][KC][TILE];

  const float* t0 = t;
  const float* t1 = t + (size_t)S_DIM * S_DIM;

  const int tid = threadIdx.x;
  const int tx  = tid & 7;     // 8 column groups x 4 cols
  const int ty  = tid >> 3;    // 32 rows
  const int c0  = blockIdx.x * TILE;
  const int r0  = blockIdx.y * TILE;

  float acc0[4] = {0.f, 0.f, 0.f, 0.f};   // p=0 accumulators (max of mins)
  float acc1[4] = {0.f, 0.f, 0.f, 0.f};   // p=1 accumulators

  // ---- prologue: async-stage chunk 0 into buffer 0 (6 b128 per thread) ----
  {
    const int k0 = 0, buf = 0;
#pragma unroll
    for (int e = 0; e < 2; ++e) {
      const int f4 = tid + e * 256;          // 0..511 float4 slots
      const int sl = f4 >> 4, kg = f4 & 15;  // A: 32 rows x 16 float4
      async_ld_b128(lds_off(&Ash[buf][sl][kg * 4]),
                    t0 + (size_t)(r0 + sl) * S_DIM + (k0 + kg * 4));
      const int kl = f4 >> 3, og = f4 & 7;   // B: 64 rows x 8 float4
      async_ld_b128(lds_off(&B0sh[buf][kl][og * 4]),
                    t0 + (size_t)(k0 + kl) * S_DIM + (c0 + og * 4));
      async_ld_b128(lds_off(&B1sh[buf][kl][og * 4]),
                    t1 + (size_t)(k0 + kl) * S_DIM + (c0 + og * 4));
    }
  }

  const int NCHUNK = S_DIM / KC;  // 8
  for (int c = 0; c < NCHUNK; ++c) {
    const int cur = c & 1;

    if (c + 1 < NCHUNK) {
      // prefetch next chunk into the other buffer (safe: end-of-iter barrier
      // of the previous iteration guarantees nobody still reads it)
      const int k0 = (c + 1) * KC, buf = cur ^ 1;
#pragma unroll
      for (int e = 0; e < 2; ++e) {
        const int f4 = tid + e * 256;
        const int sl = f4 >> 4, kg = f4 & 15;
        async_ld_b128(lds_off(&Ash[buf][sl][kg * 4]),
                      t0 + (size_t)(r0 + sl) * S_DIM + (k0 + kg * 4));
        const int kl = f4 >> 3, og = f4 & 7;
        async_ld_b128(lds_off(&B0sh[buf][kl][og * 4]),
                      t0 + (size_t)(k0 + kl) * S_DIM + (c0 + og * 4));
        async_ld_b128(lds_off(&B1sh[buf][kl][og * 4]),
                      t1 + (size_t)(k0 + kl) * S_DIM + (c0 + og * 4));
      }
      // 6 new ops in flight; async loads complete in order, so <=6 drains chunk c
      asm volatile("s_wait_asynccnt 0x6" ::: "memory");
    } else {
      asm volatile("s_wait_asynccnt 0x0" ::: "memory");
    }
    __syncthreads();

    const float* Ar  = &Ash [cur][ty][0];
    const float* B0r = &B0sh[cur][0][tx * 4];
    const float* B1r = &B1sh[cur][0][tx * 4];

#pragma unroll
    for (int k4 = 0; k4 < KC / 4; ++k4) {
      const float4 a4 = *(const float4*)(Ar + k4 * 4);
      const float av[4] = {a4.x, a4.y, a4.z, a4.w};
#pragma unroll
      for (int j = 0; j < 4; ++j) {
        const float  a  = av[j];
        const float4 b0 = *(const float4*)(B0r + (k4 * 4 + j) * TILE);
        const float4 b1 = *(const float4*)(B1r + (k4 * 4 + j) * TILE);
        acc0[0] = fmaxf(acc0[0], fminf(a, b0.x));
        acc0[1] = fmaxf(acc0[1], fminf(a, b0.y));
        acc0[2] = fmaxf(acc0[2], fminf(a, b0.z));
        acc0[3] = fmaxf(acc0[3], fminf(a, b0.w));
        acc1[0] = fmaxf(acc1[0], fminf(a, b1.x));
        acc1[1] = fmaxf(acc1[1], fminf(a, b1.y));
        acc1[2] = fmaxf(acc1[2], fminf(a, b1.z));
        acc1[3] = fmaxf(acc1[3], fminf(a, b1.w));
      }
    }
    __syncthreads();
  }

  // ---- epilogue: out = t + num - t*num, both planes, b128 stores ----
  const size_t so = (size_t)(r0 + ty) * S_DIM + (c0 + tx * 4);
  const float4 x0 = *(const float4*)(t0 + so);
  const float4 x1 = *(const float4*)(t1 + so);
  float4 o0, o1;
  o0.x = x0.x + acc0[0] - x0.x * acc0[0];
  o0.y = x0.y + acc0[1] - x0.y * acc0[1];
  o0.z = x0.z + acc0[2] - x0.z * acc0[2];
  o0.w = x0.w + acc0[3] - x0.w * acc0[3];
  o1.x = x1.x + acc1[0] - x1.x * acc1[0];
  o1.y = x1.y + acc1[1] - x1.y * acc1[1];
  o1.z = x1.z + acc1[2] - x1.z * acc1[2];
  o1.w = x1.w + acc1[3] - x1.w * acc1[3];
  *(float4*)(out + so)                            = o0;
  *(float4*)(out + (size_t)S_DIM * S_DIM + so)    = o1;
}

extern "C" void kernel_launch(void* const* d_in, const int* in_sizes, int n_in,
                              void* d_out, int out_size, void* d_ws, size_t ws_size,
                              hipStream_t stream) {
  (void)in_sizes; (void)n_in; (void)out_size; (void)d_ws; (void)ws_size;
  const float* t = (const float*)d_in[0];
  float* out = (float*)d_out;
  dim3 grid(S_DIM / TILE, S_DIM / TILE);  // 16 x 16 = 256 workgroups
  fuzzy_maxmin_decoder<<<grid, dim3(256), 0, stream>>>(t, out);
}